// GCN_40982577938827
// MI455X (gfx1250) — compile-verified
//
#include <hip/hip_runtime.h>

#define FEATS 64

typedef __attribute__((ext_vector_type(2))) float v2f;
typedef __attribute__((ext_vector_type(8))) float v8f;

// ---------------------------------------------------------------- utilities
__global__ void gcn_zero_kernel(float* __restrict__ p, int n) {
    int i = blockIdx.x * blockDim.x + threadIdx.x;
    int stride = gridDim.x * blockDim.x;
    for (; i < n; i += stride) p[i] = 0.0f;
}

__global__ void gcn_degree_kernel(const int* __restrict__ src, const int* __restrict__ dst,
                                  float* __restrict__ deg_out, float* __restrict__ deg_in, int nE) {
    int i = blockIdx.x * blockDim.x + threadIdx.x;
    int stride = gridDim.x * blockDim.x;
    for (; i < nE; i += stride) {
        __hip_atomic_fetch_add(&deg_out[src[i]], 1.0f, __ATOMIC_RELAXED, __HIP_MEMORY_SCOPE_AGENT);
        __hip_atomic_fetch_add(&deg_in[dst[i]],  1.0f, __ATOMIC_RELAXED, __HIP_MEMORY_SCOPE_AGENT);
    }
}

__global__ void gcn_invsqrt_kernel(float* __restrict__ deg_out, float* __restrict__ deg_in, int n) {
    int i = blockIdx.x * blockDim.x + threadIdx.x;
    if (i < n) {
        deg_out[i] = rsqrtf(fmaxf(deg_out[i], 1.0f));
        deg_in[i]  = rsqrtf(fmaxf(deg_in[i],  1.0f));
    }
}

// ------------------------------------------------- fused preproc + f32 WMMA GEMM
// h_out[row, :] = preproc(in[row, :]) @ W        (W is FEATS x FEATS, row-major)
// mode 0: preproc = in * inv_o[row]                         (layer 1 input = x)
// mode 1: preproc = relu(in * inv_i[row] + bprev) * inv_o   (layers 2, 3)
//
// 256 threads = 8 waves; each wave computes a 32x64 output tile (2 A-tiles
// sharing every B fragment), block covers 256 rows.
__global__ __launch_bounds__(256) void gcn_gemm_kernel(
    const float* __restrict__ in, const float* __restrict__ W,
    const float* __restrict__ bprev,
    const float* __restrict__ inv_o, const float* __restrict__ inv_i,
    float* __restrict__ h_out, int n, int mode)
{
    // W transposed: sWt[col][k] = W[k][col]  -> B fragment = contiguous float2.
    __shared__ float sWt[FEATS][FEATS + 2];
    __shared__ float sA[8][32][FEATS + 2];      // per-wave 32x64 A tile, even pad

    const int tid    = threadIdx.x;
    const int wave   = tid >> 5;
    const int lane   = tid & 31;
    const int lane16 = lane & 15;
    const int hi     = lane >> 4;               // 0: K pair {0,1}; 1: K pair {2,3}

    // Stage W transposed into LDS (4096 floats / 256 threads = 16 each).
    #pragma unroll
    for (int j = 0; j < 16; ++j) {
        int idx = tid + 256 * j;                // idx = k*64 + col
        sWt[idx & 63][idx >> 6] = W[idx];
    }

    const int rowBase = blockIdx.x * 256 + wave * 32;

    float bc0 = 0.0f, bc1 = 0.0f;
    if (mode != 0) { bc0 = bprev[2 * lane]; bc1 = bprev[2 * lane + 1]; }

    // Stage the 32x64 A tile with the fused element-wise preprocessing.
    for (int r = 0; r < 32; ++r) {
        const int row = rowBase + r;
        const int c0  = 2 * lane;
        float vx = 0.0f, vy = 0.0f;
        if (row < n) {
            const float2 raw = *(const float2*)(in + (size_t)row * FEATS + c0);
            if (mode == 0) {
                const float so = inv_o[row];
                vx = raw.x * so;
                vy = raw.y * so;
            } else {
                const float si = inv_i[row];
                const float so = inv_o[row];
                vx = fmaxf(raw.x * si + bc0, 0.0f) * so;
                vy = fmaxf(raw.y * si + bc1, 0.0f) * so;
            }
        }
        sA[wave][r][c0]     = vx;
        sA[wave][r][c0 + 1] = vy;
    }
    __syncthreads();

    // D(32x64) = A(32x64) * W(64x64) via V_WMMA_F32_16X16X4_F32.
    v8f accA[4] = {{}, {}, {}, {}};   // rows rowBase + 0..15
    v8f accB[4] = {{}, {}, {}, {}};   // rows rowBase + 16..31
    #pragma unroll
    for (int s = 0; s < 16; ++s) {
        const int k = 4 * s + 2 * hi;
        const v2f a0 = *(const v2f*)&sA[wave][lane16][k];
        const v2f a1 = *(const v2f*)&sA[wave][16 + lane16][k];
        #pragma unroll
        for (int t = 0; t < 4; ++t) {
            const v2f b = *(const v2f*)&sWt[16 * t + lane16][k];
            accA[t] = __builtin_amdgcn_wmma_f32_16x16x4_f32(false, a0, false, b, (short)0, accA[t], false, false);
            accB[t] = __builtin_amdgcn_wmma_f32_16x16x4_f32(false, a1, false, b, (short)0, accB[t], false, false);
        }
    }

    // C/D layout: VGPR j -> row j (lanes 0-15) / row j+8 (lanes 16-31), col = lane16.
    #pragma unroll
    for (int j = 0; j < 8; ++j) {
        const int rowA = rowBase + j + 8 * hi;
        if (rowA < n) {
            float* o = h_out + (size_t)rowA * FEATS + lane16;
            o[0]  = accA[0][j];
            o[16] = accA[1][j];
            o[32] = accA[2][j];
            o[48] = accA[3][j];
        }
        const int rowB = rowBase + 16 + j + 8 * hi;
        if (rowB < n) {
            float* o = h_out + (size_t)rowB * FEATS + lane16;
            o[0]  = accB[0][j];
            o[16] = accB[1][j];
            o[32] = accB[2][j];
            o[48] = accB[3][j];
        }
    }
}

// ---------------------------------------------------- edge gather/scatter (L2 atomics)
// One wave per edge; each lane carries 2 consecutive features (32*2 = 64).
__global__ __launch_bounds__(256) void gcn_scatter_kernel(
    const float* __restrict__ h, const int* __restrict__ src, const int* __restrict__ dst,
    float* __restrict__ agg, int nE)
{
    const int gw   = (blockIdx.x * 256 + threadIdx.x) >> 5;
    const int lane = threadIdx.x & 31;
    if (gw >= nE) return;
    const int s = src[gw];
    const int d = dst[gw];
    const float2 v = *(const float2*)(h + (size_t)s * FEATS + 2 * lane);
    float* a = agg + (size_t)d * FEATS + 2 * lane;
    __hip_atomic_fetch_add(a,     v.x, __ATOMIC_RELAXED, __HIP_MEMORY_SCOPE_AGENT);
    __hip_atomic_fetch_add(a + 1, v.y, __ATOMIC_RELAXED, __HIP_MEMORY_SCOPE_AGENT);
}

// ---------------------------------------------------------------- finalize
__global__ void gcn_finalize_kernel(const float* __restrict__ agg, const float* __restrict__ inv_i,
                                    const float* __restrict__ b, float* __restrict__ out, int n) {
    const int i = blockIdx.x * blockDim.x + threadIdx.x;
    if (i < n * FEATS) {
        const int row = i >> 6;
        const int col = i & 63;
        out[i] = agg[i] * inv_i[row] + b[col];
    }
}

// ---------------------------------------------------------------- launcher
extern "C" void kernel_launch(void* const* d_in, const int* in_sizes, int n_in,
                              void* d_out, int out_size, void* d_ws, size_t ws_size,
                              hipStream_t stream) {
    (void)n_in; (void)out_size; (void)ws_size;

    const float* x    = (const float*)d_in[0];
    const int*   esrc = (const int*)d_in[1];
    const int*   edst = (const int*)d_in[2];
    const float* W1   = (const float*)d_in[3];
    const float* b1   = (const float*)d_in[4];
    const float* W2   = (const float*)d_in[5];
    const float* b2   = (const float*)d_in[6];
    const float* W3   = (const float*)d_in[7];
    const float* b3   = (const float*)d_in[8];
    float* out = (float*)d_out;

    const int n  = in_sizes[0] / FEATS;   // 50000
    const int nE = in_sizes[1];           // 800000

    float* ws      = (float*)d_ws;
    float* inv_o   = ws;                                // n  (deg_out -> inv_sqrt in place)
    float* inv_i   = ws + n;                            // n
    float* hproj   = ws + 2 * (size_t)n;                // n*64
    float* agg     = ws + 2 * (size_t)n + (size_t)n * FEATS;  // n*64

    const int gemmBlocks = (n + 255) / 256;
    const int scatBlocks = (int)(((long long)nE * 32 + 255) / 256);
    const int zeroBlocksAgg = 2048;

    // Degrees -> inv-sqrt scalings (shared by all three layers).
    gcn_zero_kernel<<<512, 256, 0, stream>>>(inv_o, 2 * n);
    gcn_degree_kernel<<<(nE + 255) / 256, 256, 0, stream>>>(esrc, edst, inv_o, inv_i, nE);
    gcn_invsqrt_kernel<<<(n + 255) / 256, 256, 0, stream>>>(inv_o, inv_i, n);

    // Layer 1: h = (x * inv_o) @ W1 ; agg = scatter-sum
    gcn_gemm_kernel<<<gemmBlocks, 256, 0, stream>>>(x, W1, b1, inv_o, inv_i, hproj, n, 0);
    gcn_zero_kernel<<<zeroBlocksAgg, 256, 0, stream>>>(agg, n * FEATS);
    gcn_scatter_kernel<<<scatBlocks, 256, 0, stream>>>(hproj, esrc, edst, agg, nE);

    // Layer 2: h = (relu(agg*inv_i + b1) * inv_o) @ W2 ; agg = scatter-sum
    gcn_gemm_kernel<<<gemmBlocks, 256, 0, stream>>>(agg, W2, b1, inv_o, inv_i, hproj, n, 1);
    gcn_zero_kernel<<<zeroBlocksAgg, 256, 0, stream>>>(agg, n * FEATS);
    gcn_scatter_kernel<<<scatBlocks, 256, 0, stream>>>(hproj, esrc, edst, agg, nE);

    // Layer 3: h = (relu(agg*inv_i + b2) * inv_o) @ W3 ; agg = scatter-sum
    gcn_gemm_kernel<<<gemmBlocks, 256, 0, stream>>>(agg, W3, b2, inv_o, inv_i, hproj, n, 1);
    gcn_zero_kernel<<<zeroBlocksAgg, 256, 0, stream>>>(agg, n * FEATS);
    gcn_scatter_kernel<<<scatBlocks, 256, 0, stream>>>(hproj, esrc, edst, agg, nE);

    // Output: out = agg * inv_i + b3
    gcn_finalize_kernel<<<(n * FEATS + 255) / 256, 256, 0, stream>>>(agg, inv_i, b3, out, n);
}